// Attention_120259084452
// MI455X (gfx1250) — compile-verified
//
#include <hip/hip_runtime.h>
#include <math.h>

// ---------------------------------------------------------------------------
// SAGAN self-attention block on MI455X (gfx1250, wave32).
// All GEMMs use V_WMMA_F32_16X16X4_F32 (fp32 matrix pipe) to match the f32
// reference. Attention is fused (2-pass online softmax per 16-row s-tile) so
// the 268MB beta tensor is never materialized; HBM traffic ~150MB.
// Workspace layout (floats):
//   wcat  [96][128]  (spectral-normalized Wtheta|Wphi|Wg stacked)
//   wo_s  [128][64]
//   yfull [16][96][4096]   full-res projections (rows 0-15 = theta)
//   phi   [16][16][1024]   pooled phi
//   gT    [16][1024][64]   pooled g, transposed for coalesced WMMA-A loads
//   obuf  [16][64][4096]   attention output
// Total ~47.3MB.
// ---------------------------------------------------------------------------

typedef __attribute__((ext_vector_type(2))) float v2f;
typedef __attribute__((ext_vector_type(8))) float v8f;

// D = A(16x4) * B(4x16) + C, fp32. 8-arg form: (neg_a, A, neg_b, B, c_mod, C, reuse_a, reuse_b)
#define WMMA_F32(a, b, c) \
  __builtin_amdgcn_wmma_f32_16x16x4_f32(false, (a), false, (b), (short)0, (c), false, false)

// ---------------------------------------------------------------------------
// Kernel 0: spectral normalization.  sv = || normalize(u @ W) @ W^T ||.
// One 128-thread block per weight (gridDim.x == 4). Scaled weights written to
// wcat (theta rows 0-15, phi 16-31, g 32-95) and wo_s.
// ---------------------------------------------------------------------------
__global__ void sn_kernel(const float* wt, const float* wp, const float* wg, const float* wo,
                          const float* ut, const float* up, const float* ug, const float* uo,
                          float* wcat, float* wo_s) {
  __shared__ float v[128];
  __shared__ float red[128];
  const int wi = blockIdx.x;
  const float* W; const float* u; float* dst; int out, in;
  if (wi == 0)      { W = wt; u = ut; dst = wcat;          out = 16;  in = 128; }
  else if (wi == 1) { W = wp; u = up; dst = wcat + 16*128; out = 16;  in = 128; }
  else if (wi == 2) { W = wg; u = ug; dst = wcat + 32*128; out = 64;  in = 128; }
  else              { W = wo; u = uo; dst = wo_s;          out = 128; in = 64;  }
  const int tid = threadIdx.x;

  // v_raw = u @ W   (length `in`)
  float val = 0.f;
  if (tid < in) {
    for (int o = 0; o < out; ++o) val += u[o] * W[o * in + tid];
  }
  red[tid] = val * val;
  __syncthreads();
  for (int s = 64; s > 0; s >>= 1) { if (tid < s) red[tid] += red[tid + s]; __syncthreads(); }
  float nrm = sqrtf(red[0]);
  v[tid] = (tid < in) ? (val / fmaxf(nrm, 1e-12f)) : 0.f;
  __syncthreads();

  // sv = || v @ W^T ||
  float val2 = 0.f;
  if (tid < out) {
    for (int i = 0; i < in; ++i) val2 += v[i] * W[tid * in + i];
  }
  __syncthreads();
  red[tid] = val2 * val2;
  __syncthreads();
  for (int s = 64; s > 0; s >>= 1) { if (tid < s) red[tid] += red[tid + s]; __syncthreads(); }
  const float inv = 1.f / sqrtf(red[0]);
  for (int e = tid; e < out * in; e += blockDim.x) dst[e] = W[e] * inv;
}

// ---------------------------------------------------------------------------
// Kernel 1: fused projections Y[96 x 4096] = Wcat(96x128) @ x_b(128x4096).
// 192 threads = 6 waves; wave w owns M-tile rows [16w,16w+16). K=128 as 32
// WMMA k=4 steps. B loads (x) are lane-coalesced; A loads (weights) hit cache.
// ---------------------------------------------------------------------------
__global__ void proj_kernel(const float* __restrict__ x, const float* __restrict__ wcat,
                            float* __restrict__ yfull) {
  const int tid = threadIdx.x;
  const int wv = tid >> 5, ln = tid & 31, l15 = ln & 15, half = ln >> 4;
  const int b  = blockIdx.y;
  const int n0 = blockIdx.x * 16;
  const int m0 = wv * 16;
  const float* xb = x + (size_t)b * 128 * 4096;
  v8f acc = {};
  for (int ks = 0; ks < 32; ++ks) {
    const int kb = ks * 4 + half * 2;        // A/B K split across lane halves
    v2f A, Bv;
    A.x  = wcat[(m0 + l15) * 128 + kb + 0];
    A.y  = wcat[(m0 + l15) * 128 + kb + 1];
    Bv.x = xb[(size_t)(kb + 0) * 4096 + n0 + l15];
    Bv.y = xb[(size_t)(kb + 1) * 4096 + n0 + l15];
    acc = WMMA_F32(A, Bv, acc);
  }
  float* yb = yfull + (size_t)b * 96 * 4096;
#pragma unroll
  for (int r = 0; r < 8; ++r)
    yb[(size_t)(m0 + r + half * 8) * 4096 + n0 + l15] = acc[r];
}

// ---------------------------------------------------------------------------
// Kernel 2: 2x2 maxpool. phi channels (Y rows 16-31) -> phi[b][c][t];
// g channels (Y rows 32-95) -> gT[b][t][c] (transposed).
// ---------------------------------------------------------------------------
__global__ void pool_kernel(const float* __restrict__ yfull, float* __restrict__ phi,
                            float* __restrict__ gT) {
  const int PHI_TOT = 16 * 16 * 1024;
  const int G_TOT   = 16 * 64 * 1024;
  const int idx = blockIdx.x * blockDim.x + threadIdx.x;
  if (idx >= PHI_TOT + G_TOT) return;
  int b, ch, t;
  const bool isg = idx >= PHI_TOT;
  if (!isg) { b = idx >> 14; int r = idx & 16383; ch = r >> 10; t = r & 1023; }
  else      { int gi = idx - PHI_TOT; ch = gi & 63; t = (gi >> 6) & 1023; b = gi >> 16; }
  const int ph = t >> 5, pw = t & 31;
  const int srcch = isg ? (32 + ch) : (16 + ch);
  const float* src = yfull + ((size_t)b * 96 + srcch) * 4096 + (size_t)(2 * ph) * 64 + 2 * pw;
  const float m = fmaxf(fmaxf(src[0], src[1]), fmaxf(src[64], src[65]));
  if (!isg) phi[((size_t)b * 16 + ch) * 1024 + t] = m;
  else      gT[((size_t)b * 1024 + t) * 64 + ch] = m;
}

// ---------------------------------------------------------------------------
// Kernel 3: fused attention.  One block per (batch, 16-row s-tile); 4 waves.
// Pass 1: S = theta^T phi tile-by-tile (WMMA), online row max / exp-sum via
//         16-lane shuffles; waves partition the 64 t-tiles, stats merged in LDS.
// Pass 2: recompute S, P = exp(S-m)/l, transpose each 16x16 P tile through a
//         padded wave-private LDS buffer into WMMA-B layout, accumulate
//         O[64x16] = g @ P^T (4 c-tiles per wave); wave partials reduced in LDS.
// ---------------------------------------------------------------------------
__global__ void attn_kernel(const float* __restrict__ yfull, const float* __restrict__ phi,
                            const float* __restrict__ gT, float* __restrict__ obuf) {
  __shared__ float sPt[4][16 * 17];   // wave-private transposed P tile (padded: no bank conflicts)
  __shared__ float sm[4][16], sl[4][16];
  __shared__ float fm[16], fli[16];
  __shared__ float ored[4][64 * 16];  // per-wave partial O
  const int tid = threadIdx.x;
  const int wv = tid >> 5, ln = tid & 31, l15 = ln & 15, half = ln >> 4;
  const int b  = blockIdx.y;
  const int s0 = blockIdx.x * 16;
  const float* thb = yfull + (size_t)b * 96 * 4096;   // theta = rows 0..15
  const float* phb = phi + (size_t)b * 16 * 1024;

  // A tile (theta^T: M=s, K=c), loaded once, reused in both passes.
  float aA[8];
#pragma unroll
  for (int ks = 0; ks < 4; ++ks)
#pragma unroll
    for (int j = 0; j < 2; ++j) {
      const int c = ks * 4 + half * 2 + j;
      aA[ks * 2 + j] = thb[(size_t)c * 4096 + s0 + l15];
    }

  // ---- pass 1: online softmax stats over this wave's 16 t-tiles ----
  float mrun[8], lrun[8];
#pragma unroll
  for (int r = 0; r < 8; ++r) { mrun[r] = -INFINITY; lrun[r] = 0.f; }
  for (int tt = wv * 16; tt < wv * 16 + 16; ++tt) {
    const int t0 = tt * 16;
    v8f d = {};
#pragma unroll
    for (int ks = 0; ks < 4; ++ks) {
      const int c0 = ks * 4 + half * 2;
      v2f A, Bv;
      A.x  = aA[ks * 2 + 0];  A.y = aA[ks * 2 + 1];
      Bv.x = phb[(size_t)(c0 + 0) * 1024 + t0 + l15];
      Bv.y = phb[(size_t)(c0 + 1) * 1024 + t0 + l15];
      d = WMMA_F32(A, Bv, d);
    }
#pragma unroll
    for (int r = 0; r < 8; ++r) {          // row s = r + 8*half, cols = lanes
      float tm = d[r];
#pragma unroll
      for (int mk = 1; mk < 16; mk <<= 1) tm = fmaxf(tm, __shfl_xor(tm, mk, 32));
      const float mnew = fmaxf(mrun[r], tm);
      float ts = __expf(d[r] - mnew);
#pragma unroll
      for (int mk = 1; mk < 16; mk <<= 1) ts += __shfl_xor(ts, mk, 32);
      lrun[r] = lrun[r] * __expf(mrun[r] - mnew) + ts;
      mrun[r] = mnew;
    }
  }
  if (l15 == 0) {
#pragma unroll
    for (int r = 0; r < 8; ++r) { sm[wv][r + half * 8] = mrun[r]; sl[wv][r + half * 8] = lrun[r]; }
  }
  __syncthreads();
  if (tid < 16) {
    float M = sm[0][tid];
    for (int w = 1; w < 4; ++w) M = fmaxf(M, sm[w][tid]);
    float L = 0.f;
    for (int w = 0; w < 4; ++w) L += sl[w][tid] * __expf(sm[w][tid] - M);
    fm[tid] = M; fli[tid] = 1.f / L;
  }
  __syncthreads();

  // ---- pass 2: recompute S, form normalized P, accumulate O = g @ P^T ----
  v8f oacc[4] = {};
  const float* gb = gT + (size_t)b * 1024 * 64;
  for (int tt = wv * 16; tt < wv * 16 + 16; ++tt) {
    const int t0 = tt * 16;
    v8f d = {};
#pragma unroll
    for (int ks = 0; ks < 4; ++ks) {
      const int c0 = ks * 4 + half * 2;
      v2f A, Bv;
      A.x  = aA[ks * 2 + 0];  A.y = aA[ks * 2 + 1];
      Bv.x = phb[(size_t)(c0 + 0) * 1024 + t0 + l15];
      Bv.y = phb[(size_t)(c0 + 1) * 1024 + t0 + l15];
      d = WMMA_F32(A, Bv, d);
    }
    // P tile -> wave-private LDS, [s][t] layout (store coalesced, read conflict-free)
#pragma unroll
    for (int r = 0; r < 8; ++r) {
      const int srow = r + half * 8;
      sPt[wv][srow * 17 + l15] = __expf(d[r] - fm[srow]) * fli[srow];
    }
    asm volatile("s_wait_dscnt 0" ::: "memory");   // wave-local store->load ordering
#pragma unroll
    for (int ks = 0; ks < 4; ++ks) {
      const int tb = ks * 4 + half * 2;
      v2f Bv;                                        // B: K=t, N=s  (P^T)
      Bv.x = sPt[wv][l15 * 17 + tb + 0];
      Bv.y = sPt[wv][l15 * 17 + tb + 1];
#pragma unroll
      for (int ct = 0; ct < 4; ++ct) {
        const int c0 = ct * 16;
        v2f A;                                       // A: M=c, K=t (from gT: coalesced)
        A.x = gb[(size_t)(t0 + tb + 0) * 64 + c0 + l15];
        A.y = gb[(size_t)(t0 + tb + 1) * 64 + c0 + l15];
        oacc[ct] = WMMA_F32(A, Bv, oacc[ct]);
      }
    }
  }

  // reduce partial O across the 4 waves (they partition t)
#pragma unroll
  for (int ct = 0; ct < 4; ++ct)
#pragma unroll
    for (int r = 0; r < 8; ++r)
      ored[wv][(ct * 16 + r + half * 8) * 16 + l15] = oacc[ct][r];
  __syncthreads();
  float* ob = obuf + (size_t)b * 64 * 4096;
  for (int e = tid; e < 64 * 16; e += 128) {
    const float s = ored[0][e] + ored[1][e] + ored[2][e] + ored[3][e];
    const int c = e >> 4, sc = e & 15;
    ob[(size_t)c * 4096 + s0 + sc] = s;
  }
}

// ---------------------------------------------------------------------------
// Kernel 4: out = gamma * (Wo @ O) + x.  256 threads = 8 waves, one M-tile each.
// ---------------------------------------------------------------------------
__global__ void out_kernel(const float* __restrict__ obuf, const float* __restrict__ wo_s,
                           const float* __restrict__ x, const float* __restrict__ gamma_p,
                           float* __restrict__ out) {
  const int tid = threadIdx.x;
  const int wv = tid >> 5, ln = tid & 31, l15 = ln & 15, half = ln >> 4;
  const int b  = blockIdx.y;
  const int n0 = blockIdx.x * 16;
  const int m0 = wv * 16;
  const float* ob = obuf + (size_t)b * 64 * 4096;
  v8f acc = {};
  for (int ks = 0; ks < 16; ++ks) {
    const int kb = ks * 4 + half * 2;
    v2f A, Bv;
    A.x  = wo_s[(m0 + l15) * 64 + kb + 0];
    A.y  = wo_s[(m0 + l15) * 64 + kb + 1];
    Bv.x = ob[(size_t)(kb + 0) * 4096 + n0 + l15];
    Bv.y = ob[(size_t)(kb + 1) * 4096 + n0 + l15];
    acc = WMMA_F32(A, Bv, acc);
  }
  const float g = gamma_p[0];
  const float* xb  = x   + (size_t)b * 128 * 4096;
  float*       onb = out + (size_t)b * 128 * 4096;
#pragma unroll
  for (int r = 0; r < 8; ++r) {
    const size_t off = (size_t)(m0 + r + half * 8) * 4096 + n0 + l15;
    onb[off] = g * acc[r] + xb[off];
  }
}

// ---------------------------------------------------------------------------
extern "C" void kernel_launch(void* const* d_in, const int* in_sizes, int n_in,
                              void* d_out, int out_size, void* d_ws, size_t ws_size,
                              hipStream_t stream) {
  const float* x  = (const float*)d_in[0];
  const float* wt = (const float*)d_in[1];
  const float* wp = (const float*)d_in[2];
  const float* wg = (const float*)d_in[3];
  const float* wo = (const float*)d_in[4];
  const float* ut = (const float*)d_in[5];
  const float* up = (const float*)d_in[6];
  const float* ug = (const float*)d_in[7];
  const float* uo = (const float*)d_in[8];
  const float* gm = (const float*)d_in[9];
  float* out = (float*)d_out;

  float* ws    = (float*)d_ws;
  float* wcat  = ws;                                   //  96*128
  float* wo_s  = wcat + 96 * 128;                      // 128*64
  float* yfull = wo_s + 128 * 64;                      // 16*96*4096
  float* phi   = yfull + (size_t)16 * 96 * 4096;       // 16*16*1024
  float* gT    = phi   + (size_t)16 * 16 * 1024;       // 16*1024*64
  float* obuf  = gT    + (size_t)16 * 1024 * 64;       // 16*64*4096

  sn_kernel<<<4, 128, 0, stream>>>(wt, wp, wg, wo, ut, up, ug, uo, wcat, wo_s);
  proj_kernel<<<dim3(256, 16), 192, 0, stream>>>(x, wcat, yfull);
  const int tot = 16 * 16 * 1024 + 16 * 64 * 1024;
  pool_kernel<<<(tot + 255) / 256, 256, 0, stream>>>(yfull, phi, gT);
  attn_kernel<<<dim3(256, 16), 128, 0, stream>>>(yfull, phi, gT, obuf);
  out_kernel<<<dim3(256, 16), 256, 0, stream>>>(obuf, wo_s, x, gm, out);
}